// PostGammaDenoise_60833916781014
// MI455X (gfx1250) — compile-verified
//
#include <hip/hip_runtime.h>

typedef __attribute__((ext_vector_type(2))) float v2f;
typedef __attribute__((ext_vector_type(8))) float v8f;

#define Hs 4096
#define Ws 4096
#define TX 64      // output tile width
#define TY 32      // output tile height
#define RX 96      // region width  (TX + 2*16)
#define RY 64      // region height (TY + 2*16)
#define RS 97      // LDS row stride (padded, floats)
#define KS 17
#define INV17 (1.0f / 17.0f)
#define EPSF 0.005f

__device__ __forceinline__ int reflect_idx(int c, int n) {
    c = (c < 0) ? -c : c;
    c = (c >= n) ? (2 * n - 2 - c) : c;
    return c;
}

// In-place horizontal 17-tap box over one LDS row.
// Input: row[0 .. win+15], Output: row[0 .. win-1], out[j] = mean(in[j..j+16]).
__device__ __forceinline__ void hbox_row(float* row, int win) {
    float s = 0.0f;
#pragma unroll
    for (int c = 0; c < KS; ++c) s += row[c];
    for (int j = 0; j < win; ++j) {
        float sub = row[j];          // read original before overwrite
        row[j] = s * INV17;
        if (j + 1 < win) s += row[j + KS] - sub;
    }
}

// Vertical 17-tap box of one 16x16 output chunk via f32 WMMA:
// D(16x16) = Wv(16x32, banded 1/17) x In(32x16), accumulated over 8 K=4 chunks.
// wv[] holds the lane-resident A-matrix chunks (precomputed once per kernel).
// Loads are batched into b[8] first so the DS waits amortize and the 8 WMMAs
// issue as a pure accumulate chain on the matrix pipe.
__device__ __forceinline__ v8f vbox_wmma(const float* buf, int rbase, int c0,
                                         int lane, const v2f* wv) {
    const int khalf = (lane >> 4) << 1;       // lanes 0-15 -> K0/K1, lanes 16-31 -> K2/K3
    const int col   = c0 + (lane & 15);       // B: row striped across lanes within a VGPR
    const float* p  = buf + (size_t)(rbase + khalf) * RS + col;
    v2f b[8];
#pragma unroll
    for (int k = 0; k < 8; ++k) {
        b[k].x = p[(4 * k)     * RS];
        b[k].y = p[(4 * k + 1) * RS];
    }
    v8f acc = {};
#pragma unroll
    for (int k = 0; k < 8; ++k) {
        acc = __builtin_amdgcn_wmma_f32_16x16x4_f32(
            /*neg_a=*/false, wv[k], /*neg_b=*/false, b[k],
            /*c_mod=*/(short)0, acc, /*reuse_a=*/false, /*reuse_b=*/false);
    }
    return acc;
}

__global__ __launch_bounds__(256)
void guided_denoise_kernel(const float* __restrict__ x, float* __restrict__ out) {
    __shared__ float bufA[RY * RS];   // Y   -> hY   -> meanI  -> b -> hb -> mean_b
    __shared__ float bufB[RY * RS];   // Y*Y -> hYY  -> meanII -> a -> ha -> mean_a

    const int tid  = threadIdx.x;
    const int lane = tid & 31;
    const int wave = tid >> 5;
    const int gx0  = blockIdx.x * TX;
    const int gy0  = blockIdx.y * TY;

    // ---- Precompute banded A-matrix Wv chunks once (task-invariant, lane-resident) ----
    // Wv[m][r] = 1/17 iff m <= r <= m+16; A 16x4 layout: lanes 0-15 hold K0/K1, 16-31 hold K2/K3.
    v2f wv[8];
    {
        const int mrow  = lane & 15;
        const int khalf = (lane >> 4) << 1;
#pragma unroll
        for (int k = 0; k < 8; ++k) {
            const int r0 = 4 * k + khalf;
            wv[k].x = (r0     >= mrow && r0     <= mrow + 16) ? INV17 : 0.0f;
            wv[k].y = (r0 + 1 >= mrow && r0 + 1 <= mrow + 16) ? INV17 : 0.0f;
        }
    }

    // ---- Stage L: load region (reflect-padded), compute luma Y and Y^2 ----
    for (int idx = tid; idx < RX * RY; idx += 256) {
        const int r  = idx / RX;
        const int c  = idx % RX;
        const int gx = reflect_idx(gx0 - 16 + c, Ws);
        const int gy = reflect_idx(gy0 - 16 + r, Hs);
        const float* p = x + ((size_t)gy * Ws + gx) * 3;
        const float Y = 0.2126f * p[0] + 0.7152f * p[1] + 0.0722f * p[2];
        bufA[r * RS + c] = Y;
        bufB[r * RS + c] = Y * Y;
    }
    __syncthreads();

    // ---- Stage H1: horizontal box, 64 rows x 2 planes, in place; 80 outputs ----
    if (tid < 128) {
        float* row = ((tid & 1) ? bufB : bufA) + (tid >> 1) * RS;
        hbox_row(row, 80);
    }
    __syncthreads();

    // ---- Stage V1 (WMMA): out rows 0..47 (3 chunks) x cols 0..79 (5 chunks) x 2 planes = 30 tasks ----
    {
        v8f d[4];
        int tk[4];
#pragma unroll
        for (int i = 0; i < 4; ++i) {
            const int t = wave + i * 8;
            tk[i] = t;
            const int tt   = (t < 30) ? t : 0;     // clamped dummy work keeps EXEC all-1s
            const int bsel = tt & 1;
            const int rest = tt >> 1;
            const int mt   = rest % 3;
            const int cc   = rest / 3;
            const float* buf = bsel ? bufB : bufA;
            d[i] = vbox_wmma(buf, 16 * mt, cc * 16, lane, wv);
        }
        __syncthreads();   // all reads done before any in-place writes
#pragma unroll
        for (int i = 0; i < 4; ++i) {
            const int t = tk[i];
            if (t < 30) {
                const int bsel = t & 1;
                const int rest = t >> 1;
                const int mt   = rest % 3;
                const int cc   = rest / 3;
                float* buf = bsel ? bufB : bufA;
                const int row = 16 * mt + ((lane >> 4) << 3);   // D: M 0-7 / 8-15 across half-waves
                const int col = cc * 16 + (lane & 15);
#pragma unroll
                for (int j = 0; j < 8; ++j)
                    buf[(row + j) * RS + col] = d[i][j];
            }
        }
    }
    __syncthreads();

    // ---- Stage AB: a = var/(var+eps), b = meanI - a*meanI, on 48x80 grid ----
    for (int idx = tid; idx < 48 * 80; idx += 256) {
        const int r = idx / 80;
        const int c = idx % 80;
        const float mi  = bufA[r * RS + c];
        const float mii = bufB[r * RS + c];
        const float var = mii - mi * mi;
        const float a   = var / (var + EPSF);
        const float b   = mi - a * mi;
        bufB[r * RS + c] = a;
        bufA[r * RS + c] = b;
    }
    __syncthreads();

    // ---- Stage H2: horizontal box, 48 rows x 2 planes, in place; 64 outputs ----
    if (tid < 96) {
        float* row = ((tid & 1) ? bufB : bufA) + (tid >> 1) * RS;
        hbox_row(row, 64);
    }
    __syncthreads();

    // ---- Stage V2 (WMMA): out rows 0..31 (2 chunks) x cols 0..63 (4 chunks) x 2 planes = 16 tasks ----
    {
        v8f d[2];
#pragma unroll
        for (int i = 0; i < 2; ++i) {
            const int t    = wave + i * 8;
            const int bsel = t & 1;
            const int mt   = (t >> 1) & 1;
            const int cc   = t >> 2;
            const float* buf = bsel ? bufB : bufA;
            d[i] = vbox_wmma(buf, 16 * mt, cc * 16, lane, wv);
        }
        __syncthreads();
#pragma unroll
        for (int i = 0; i < 2; ++i) {
            const int t    = wave + i * 8;
            const int bsel = t & 1;
            const int mt   = (t >> 1) & 1;
            const int cc   = t >> 2;
            float* buf = bsel ? bufB : bufA;
            const int row = 16 * mt + ((lane >> 4) << 3);
            const int col = cc * 16 + (lane & 15);
#pragma unroll
            for (int j = 0; j < 8; ++j)
                buf[(row + j) * RS + col] = d[i][j];
        }
    }
    __syncthreads();

    // ---- Final: reload center x (L2-resident), apply smooth scale, clamp ----
    for (int idx = tid; idx < TX * TY; idx += 256) {
        const int r = idx / TX;
        const int c = idx % TX;
        const size_t base = ((size_t)(gy0 + r) * Ws + (gx0 + c)) * 3;
        const float x0 = x[base + 0];
        const float x1 = x[base + 1];
        const float x2 = x[base + 2];
        const float Y  = 0.2126f * x0 + 0.7152f * x1 + 0.0722f * x2;
        const float ma = bufB[r * RS + c];   // mean_a
        const float mb = bufA[r * RS + c];   // mean_b
        const float ys = ma * Y + mb;
        const float sc = ys / fmaxf(Y, 1e-6f);
        out[base + 0] = fminf(fmaxf(x0 * sc, 0.0f), 1.0f);
        out[base + 1] = fminf(fmaxf(x1 * sc, 0.0f), 1.0f);
        out[base + 2] = fminf(fmaxf(x2 * sc, 0.0f), 1.0f);
    }
}

extern "C" void kernel_launch(void* const* d_in, const int* in_sizes, int n_in,
                              void* d_out, int out_size, void* d_ws, size_t ws_size,
                              hipStream_t stream) {
    (void)in_sizes; (void)n_in; (void)d_ws; (void)ws_size; (void)out_size;
    const float* x = (const float*)d_in[0];
    float* out = (float*)d_out;
    dim3 grid(Ws / TX, Hs / TY);   // 64 x 128 tiles
    guided_denoise_kernel<<<grid, dim3(256), 0, stream>>>(x, out);
}